// HABlock_60593398612616
// MI455X (gfx1250) — compile-verified
//
#include <hip/hip_runtime.h>
#include <math.h>
#include <stdint.h>

#define HD 256
#define WD 256
#define CCH 96
#define NHEAD 6
#define DHEAD 16
#define WSZ 8
#define SHIFT_ 4
#define NPIX 65536              // H*W
#define NTOT (2 * CCH * NPIX)   // full tensor elements

typedef __attribute__((ext_vector_type(2))) float v2f;
typedef __attribute__((ext_vector_type(8))) float v8f;

__device__ __forceinline__ v8f wmma_f32(v2f a, v2f b, v8f c) {
  // D = A(16x4) * B(4x16) + C, fp32
  return __builtin_amdgcn_wmma_f32_16x16x4_f32(false, a, false, b, (short)0, c,
                                               false, false);
}

__device__ __forceinline__ float gelu_exact(float x) {
  return 0.5f * x * (1.0f + erff(x * 0.70710678118654752f));
}

// ---- CDNA5 async global->LDS copy (guarded: falls back to plain load) ----
#if __has_builtin(__builtin_amdgcn_global_load_async_to_lds_b32)
#define ASYNC_LDS 1
__device__ __forceinline__ void async_copy_b32(const float* g, float* l) {
  __builtin_amdgcn_global_load_async_to_lds_b32(
      (__attribute__((address_space(1))) int*)(uintptr_t)g,
      (__attribute__((address_space(3))) int*)(uint32_t)(uintptr_t)l,
      0, 0);
}
__device__ __forceinline__ void async_wait0() {
#if __has_builtin(__builtin_amdgcn_s_wait_asynccnt)
  __builtin_amdgcn_s_wait_asynccnt(0);
#else
  asm volatile("s_wait_asynccnt 0" ::: "memory");
#endif
}
#endif

// ---------------------------------------------------------------- transpose
// Build row-major W^T (and zero-padded conv weights) so WMMA B-fragments load
// coalesced: BT[k][n] = W[n][k].  Conv K-order is (tap, ic).
__global__ void k_transpose(const float* qkv_w, const float* proj_w,
                            const float* mlp1_w, const float* mlp2_w,
                            const float* cab1_w, const float* cab2_w,
                            float* qkvT, float* projT, float* m1T, float* m2T,
                            float* cw1T, float* cw2T) {
  int i = blockIdx.x * blockDim.x + threadIdx.x;
  if (i < 96 * 288) { int k = i / 288, n = i % 288; qkvT[i] = qkv_w[n * 96 + k]; }
  if (i < 96 * 96)  { int k = i / 96,  n = i % 96;  projT[i] = proj_w[n * 96 + k]; }
  if (i < 96 * 384) { int k = i / 384, n = i % 384; m1T[i] = mlp1_w[n * 96 + k]; }
  if (i < 384 * 96) { int k = i / 96,  n = i % 96;  m2T[i] = mlp2_w[n * 384 + k]; }
  if (i < 864 * 32) {            // conv1: K = tap*96+ic, N padded 24->32
    int kk = i >> 5, oc = i & 31;
    int tap = kk / 96, ic = kk - tap * 96;
    cw1T[i] = (oc < 24) ? cab1_w[(oc * 96 + ic) * 9 + tap] : 0.f;
  }
  if (i < 216 * 96) {            // conv2: K = tap*24+ic, N = 96
    int kk = i / 96, oc = i - kk * 96;
    int tap = kk / 24, ic = kk - tap * 24;
    cw2T[i] = cab2_w[(oc * 24 + ic) * 9 + tap];
  }
}

// ---------------------------------------------------------------- channel LN
__global__ void k_ln(const float* __restrict__ x, const float* __restrict__ w,
                     const float* __restrict__ b, float* __restrict__ y) {
  int p = blockIdx.x * blockDim.x + threadIdx.x;  // 0 .. B*NPIX-1
  int bb = p >> 16, sp = p & (NPIX - 1);
  const float* xp = x + ((size_t)bb * CCH << 16) + sp;
  float s = 0.f, ss = 0.f;
  for (int c = 0; c < CCH; c++) {
    float v = xp[(size_t)c << 16];
    s += v; ss += v * v;
  }
  float mu = s / CCH;
  float var = ss / CCH - mu * mu;
  float rstd = rsqrtf(var + 1e-5f);
  float* yp = y + ((size_t)bb * CCH << 16) + sp;
  for (int c = 0; c < CCH; c++) {
    float v = xp[(size_t)c << 16];
    yp[(size_t)c << 16] = (v - mu) * rstd * w[c] + b[c];
  }
}

// ---------------------------------------------------------------- swin attn
#define XST 100   // padded stride for 96-wide LDS rows (bank-conflict free)
#define QST 20    // padded stride for 16-wide LDS rows
#define SST 68    // padded stride for 64-wide LDS rows

__device__ __forceinline__ int reglab(int y) {
  return (y < HD - WSZ) ? 0 : (y < HD - SHIFT_ ? 1 : 2);
}

__global__ __launch_bounds__(256) void k_swin(
    const float* __restrict__ u, const float* __restrict__ qkvT,
    const float* __restrict__ qkv_b, const float* __restrict__ projT,
    const float* __restrict__ proj_b, const float* __restrict__ rpb,
    float* __restrict__ msa) {
  __shared__ float Xs[64 * XST];
  __shared__ float Qs[64 * QST];
  __shared__ float Ks[64 * QST];
  __shared__ float Vs[64 * QST];
  __shared__ float Ss[64 * SST];
  __shared__ float Os[64 * QST];

  int tid = threadIdx.x;
  int wave = tid >> 5, lane = tid & 31;
  int lr = lane & 15;
  bool hifrag = lane >= 16;

  int wg = blockIdx.x;
  int bb = wg >> 10;               // 1024 windows per image
  int wid = wg & 1023;
  int wh = wid >> 5, ww = wid & 31;

  // Load shifted window into LDS: Xs[pixel][channel]
  for (int i = tid; i < 64 * CCH; i += 256) {
    int ch = i >> 6, row = i & 63;
    int rr = row >> 3, cc = row & 7;
    int gy = (wh * WSZ + rr + SHIFT_) & (HD - 1);
    int gx = (ww * WSZ + cc + SHIFT_) & (WD - 1);
    const float* gp = &u[(((size_t)bb * CCH + ch) << 16) + (gy << 8) + gx];
#ifdef ASYNC_LDS
    async_copy_b32(gp, &Xs[row * XST + ch]);
#else
    Xs[row * XST + ch] = *gp;
#endif
  }
#ifdef ASYNC_LDS
  async_wait0();
#endif
  __syncthreads();

  v8f outacc[3] = {};  // 24 proj-output tiles / 8 waves = 3 per wave

  for (int h = 0; h < NHEAD; h++) {
    // ---- QKV GEMM for this head: [64x96] @ 3x[96x16]
    for (int t = wave; t < 12; t += 8) {
      int which = t >> 2, mt = t & 3;
      int cb = which * CCH + h * DHEAD;  // column base in the 288-wide QKV
      v8f acc = {};
      for (int k = 0; k < CCH; k += 4) {
        int kk = k + (hifrag ? 2 : 0);
        v2f a = *(const v2f*)&Xs[(mt * 16 + lr) * XST + kk];
        v2f bv;
        bv.x = qkvT[kk * 288 + cb + lr];
        bv.y = qkvT[(kk + 1) * 288 + cb + lr];
        acc = wmma_f32(a, bv, acc);
      }
      float* dst = (which == 0) ? Qs : (which == 1 ? Ks : Vs);
#pragma unroll
      for (int r = 0; r < 8; r++) {
        int row = mt * 16 + r + (hifrag ? 8 : 0);
        dst[row * QST + lr] = acc[r] + qkv_b[cb + lr];
      }
    }
    __syncthreads();

    // ---- S = Q @ K^T  (scale + rel-pos bias + shift mask fused on store)
    for (int t = wave; t < 16; t += 8) {
      int mt = t >> 2, nt = t & 3;
      v8f acc = {};
      for (int k = 0; k < DHEAD; k += 4) {
        int kk = k + (hifrag ? 2 : 0);
        v2f a = *(const v2f*)&Qs[(mt * 16 + lr) * QST + kk];
        v2f bv = *(const v2f*)&Ks[(nt * 16 + lr) * QST + kk];  // B = K^T
        acc = wmma_f32(a, bv, acc);
      }
#pragma unroll
      for (int r = 0; r < 8; r++) {
        int n = mt * 16 + r + (hifrag ? 8 : 0);  // query index
        int m = nt * 16 + lr;                    // key index
        int ri = (n >> 3) - (m >> 3) + 7;
        int ci = (n & 7) - (m & 7) + 7;
        float bias = rpb[(ri * 15 + ci) * NHEAD + h];
        int labn = reglab(wh * WSZ + (n >> 3)) * 3 + reglab(ww * WSZ + (n & 7));
        int labm = reglab(wh * WSZ + (m >> 3)) * 3 + reglab(ww * WSZ + (m & 7));
        float maskv = (labn != labm) ? -100.0f : 0.0f;
        Ss[n * SST + m] = acc[r] * 0.25f + bias + maskv;  // 1/sqrt(16)
      }
    }
    __syncthreads();

    // ---- row softmax
    if (tid < 64) {
      float mx = -1e30f;
      for (int m2 = 0; m2 < 64; m2++) mx = fmaxf(mx, Ss[tid * SST + m2]);
      float sum = 0.f;
      for (int m2 = 0; m2 < 64; m2++) {
        float e = expf(Ss[tid * SST + m2] - mx);
        Ss[tid * SST + m2] = e;
        sum += e;
      }
      float inv = 1.0f / sum;
      for (int m2 = 0; m2 < 64; m2++) Ss[tid * SST + m2] *= inv;
    }
    __syncthreads();

    // ---- O = S @ V  ([64x64]@[64x16], 4 tiles)
    if (wave < 4) {
      int mt = wave;
      v8f acc = {};
      for (int k = 0; k < 64; k += 4) {
        int kk = k + (hifrag ? 2 : 0);
        v2f a = *(const v2f*)&Ss[(mt * 16 + lr) * SST + kk];
        v2f bv;
        bv.x = Vs[kk * QST + lr];
        bv.y = Vs[(kk + 1) * QST + lr];
        acc = wmma_f32(a, bv, acc);
      }
#pragma unroll
      for (int r = 0; r < 8; r++) {
        int row = mt * 16 + r + (hifrag ? 8 : 0);
        Os[row * QST + lr] = acc[r];
      }
    }
    __syncthreads();

    // ---- projection: accumulate O_h @ projT[h*16:(h+1)*16, :] in registers
    for (int j = 0; j < 3; j++) {
      int t = wave + j * 8;
      int mt = t & 3, nt = t >> 2;
      v8f acc = outacc[j];
      for (int k = 0; k < DHEAD; k += 4) {
        int kk = k + (hifrag ? 2 : 0);
        v2f a = *(const v2f*)&Os[(mt * 16 + lr) * QST + kk];
        v2f bv;
        bv.x = projT[(h * 16 + kk) * CCH + nt * 16 + lr];
        bv.y = projT[(h * 16 + kk + 1) * CCH + nt * 16 + lr];
        acc = wmma_f32(a, bv, acc);
      }
      outacc[j] = acc;
    }
    __syncthreads();
  }

  // ---- write MSA with inverse shift (same index map as the load)
  for (int j = 0; j < 3; j++) {
    int t = wave + j * 8;
    int mt = t & 3, nt = t >> 2;
#pragma unroll
    for (int r = 0; r < 8; r++) {
      int row = mt * 16 + r + (hifrag ? 8 : 0);
      int col = nt * 16 + lr;
      float val = outacc[j][r] + proj_b[col];
      int rr = row >> 3, cc = row & 7;
      int gy = (wh * WSZ + rr + SHIFT_) & (HD - 1);
      int gx = (ww * WSZ + cc + SHIFT_) & (WD - 1);
      msa[(((size_t)bb * CCH + col) << 16) + (gy << 8) + gx] = val;
    }
  }
}

// --------------------------------------------- CAB convs as implicit GEMM
// One WG per 16x16 spatial tile. LDS holds halo'd 18x18 input tile,
// channel-minor with padded stride (bank-conflict-free v2f A-fragments).
#define T1ST 20   // conv1 chunk: 16 channels -> pad 20
#define T2ST 28   // conv2:       24 channels -> pad 28

__global__ __launch_bounds__(256) void k_conv1(
    const float* __restrict__ u, const float* __restrict__ w1T,
    const float* __restrict__ bias, float* __restrict__ c1) {
  __shared__ float Tin[324 * T1ST];  // 18*18 x 16ch = 25.9 KB
  int tid = threadIdx.x, wave = tid >> 5, lane = tid & 31;
  int lr = lane & 15;
  bool hifrag = lane >= 16;
  int bb = blockIdx.x >> 8;
  int tile = blockIdx.x & 255;
  int ty0 = (tile >> 4) << 4, tx0 = (tile & 15) << 4;

  v8f acc[4] = {};  // 16 M-tiles x 2 N-tiles / 8 waves = 4 per wave
  for (int c6 = 0; c6 < 6; c6++) {  // 6 chunks of 16 input channels
    int ic0 = c6 * 16;
    __syncthreads();
    for (int i = tid; i < 324 * 16; i += 256) {
      int icl = i / 324, pix = i - icl * 324;
      int yy = pix / 18, xx = pix - yy * 18;
      int gy = ty0 + yy - 1, gx = tx0 + xx - 1;
      float v = 0.f;
      if (gy >= 0 && gy < HD && gx >= 0 && gx < WD)
        v = u[(((size_t)bb * CCH + ic0 + icl) << 16) + (gy << 8) + gx];
      Tin[pix * T1ST + icl] = v;
    }
    __syncthreads();
    for (int tap = 0; tap < 9; tap++) {
      int dy = tap / 3, dx = tap - dy * 3;  // halo offsets 0..2
      for (int j = 0; j < 4; j++) {
        int t = j * 8 + wave;
        int mt = t >> 1, nt = t & 1;        // mt = tile row (py), px = lr
        v8f a8 = acc[j];
        for (int k = 0; k < 16; k += 4) {
          int kk = k + (hifrag ? 2 : 0);
          v2f a = *(const v2f*)&Tin[((mt + dy) * 18 + lr + dx) * T1ST + kk];
          int krow = tap * 96 + ic0 + kk;
          v2f bv;
          bv.x = w1T[krow * 32 + nt * 16 + lr];
          bv.y = w1T[(krow + 1) * 32 + nt * 16 + lr];
          a8 = wmma_f32(a, bv, a8);
        }
        acc[j] = a8;
      }
    }
  }
  for (int j = 0; j < 4; j++) {
    int t = j * 8 + wave;
    int mt = t >> 1, nt = t & 1;
    int oc = nt * 16 + lr;
    if (oc < 24) {
#pragma unroll
      for (int r = 0; r < 8; r++) {
        int px = r + (hifrag ? 8 : 0);
        float v = gelu_exact(acc[j][r] + bias[oc]);
        c1[(((size_t)bb * 24 + oc) << 16) + ((ty0 + mt) << 8) + tx0 + px] = v;
      }
    }
  }
}

__global__ __launch_bounds__(256) void k_conv2(
    const float* __restrict__ c1, const float* __restrict__ w2T,
    const float* __restrict__ bias, float* __restrict__ c2) {
  __shared__ float Tin[324 * T2ST];  // 18*18 x 24ch = 36.3 KB
  int tid = threadIdx.x, wave = tid >> 5, lane = tid & 31;
  int lr = lane & 15;
  bool hifrag = lane >= 16;
  int bb = blockIdx.x >> 8;
  int tile = blockIdx.x & 255;
  int ty0 = (tile >> 4) << 4, tx0 = (tile & 15) << 4;

  for (int i = tid; i < 324 * 24; i += 256) {
    int icl = i / 324, pix = i - icl * 324;
    int yy = pix / 18, xx = pix - yy * 18;
    int gy = ty0 + yy - 1, gx = tx0 + xx - 1;
    float v = 0.f;
    if (gy >= 0 && gy < HD && gx >= 0 && gx < WD)
      v = c1[(((size_t)bb * 24 + icl) << 16) + (gy << 8) + gx];
    Tin[pix * T2ST + icl] = v;
  }
  __syncthreads();

  for (int j = 0; j < 12; j++) {      // 16 M-tiles x 6 N-tiles / 8 waves
    int t = j * 8 + wave;
    int mt = t & 15, nt = t >> 4;
    v8f a8 = {};
    for (int tap = 0; tap < 9; tap++) {
      int dy = tap / 3, dx = tap - dy * 3;
      for (int k = 0; k < 24; k += 4) {
        int kk = k + (hifrag ? 2 : 0);
        v2f a = *(const v2f*)&Tin[((mt + dy) * 18 + lr + dx) * T2ST + kk];
        int krow = tap * 24 + kk;
        v2f bv;
        bv.x = w2T[krow * 96 + nt * 16 + lr];
        bv.y = w2T[(krow + 1) * 96 + nt * 16 + lr];
        a8 = wmma_f32(a, bv, a8);
      }
    }
    int oc = nt * 16 + lr;
#pragma unroll
    for (int r = 0; r < 8; r++) {
      int px = r + (hifrag ? 8 : 0);
      c2[(((size_t)bb * CCH + oc) << 16) + ((ty0 + mt) << 8) + tx0 + px] =
          a8[r] + bias[oc];
    }
  }
}

// ------------------------------------------------------- channel attention
__global__ void k_pool(const float* __restrict__ c2, float* __restrict__ pool) {
  __shared__ float red[256];
  int bc = blockIdx.x;  // 0..191 (b*96+ch)
  const float* p = c2 + ((size_t)bc << 16);
  float s = 0.f;
  for (int i = threadIdx.x; i < NPIX; i += 256) s += p[i];
  red[threadIdx.x] = s;
  __syncthreads();
  for (int st = 128; st > 0; st >>= 1) {
    if (threadIdx.x < st) red[threadIdx.x] += red[threadIdx.x + st];
    __syncthreads();
  }
  if (threadIdx.x == 0) pool[bc] = red[0] / (float)NPIX;
}

__global__ void k_ca(const float* __restrict__ pool, const float* __restrict__ w1,
                     const float* __restrict__ b1, const float* __restrict__ w2,
                     const float* __restrict__ b2, float* __restrict__ ascale) {
  __shared__ float hid[6];
  int bb = blockIdx.x;
  int t = threadIdx.x;
  if (t < 6) {
    float a = b1[t];
    for (int c = 0; c < 96; c++) a += w1[t * 96 + c] * pool[bb * 96 + c];
    hid[t] = fmaxf(a, 0.f);
  }
  __syncthreads();
  if (t < 96) {
    float a = b2[t];
    for (int j = 0; j < 6; j++) a += w2[t * 6 + j] * hid[j];
    ascale[bb * 96 + t] = 1.f / (1.f + expf(-a));
  }
}

__global__ void k_combine(const float* __restrict__ x, const float* __restrict__ msa,
                          const float* __restrict__ c2, const float* __restrict__ asc,
                          float* __restrict__ xr) {
  int i = blockIdx.x * blockDim.x + threadIdx.x;
  if (i >= NTOT) return;
  int ch = (i >> 16) % 96;
  int bb = i / (96 * NPIX);
  xr[i] = msa[i] + 0.01f * c2[i] * asc[bb * 96 + ch] + x[i];
}

// ---------------------------------------------------------------- MLP (1x1)
__global__ __launch_bounds__(256) void k_mlp(
    const float* __restrict__ xr, const float* __restrict__ ln2_w,
    const float* __restrict__ ln2_b, const float* __restrict__ w1T,
    const float* __restrict__ b1, const float* __restrict__ w2T,
    const float* __restrict__ b2, float* __restrict__ out) {
  __shared__ float XN[64 * XST];
  __shared__ float Hc[64 * XST];
  int tid = threadIdx.x;
  int wave = tid >> 5, lane = tid & 31;
  int lr = lane & 15;
  bool hifrag = lane >= 16;
  size_t pix0 = (size_t)blockIdx.x * 64;
  int bb = (int)(pix0 >> 16);
  int sp = (int)(pix0 & 65535);

  // load 64 pixels x 96 channels (coalesced over pixels per channel)
  for (int i = tid; i < 64 * 96; i += 256) {
    int ch = i >> 6, row = i & 63;
    const float* gp = &xr[(((size_t)bb * 96 + ch) << 16) + sp + row];
#ifdef ASYNC_LDS
    async_copy_b32(gp, &XN[row * XST + ch]);
#else
    XN[row * XST + ch] = *gp;
#endif
  }
#ifdef ASYNC_LDS
  async_wait0();
#endif
  __syncthreads();

  // LN2 per pixel-row
  if (tid < 64) {
    float s = 0.f, ss = 0.f;
    for (int c = 0; c < 96; c++) {
      float v = XN[tid * XST + c];
      s += v; ss += v * v;
    }
    float mu = s / 96.f, var = ss / 96.f - mu * mu;
    float rstd = rsqrtf(var + 1e-5f);
    for (int c = 0; c < 96; c++)
      XN[tid * XST + c] = (XN[tid * XST + c] - mu) * rstd * ln2_w[c] + ln2_b[c];
  }
  __syncthreads();

  v8f outacc[3] = {};

  // process hidden dim (384) in 4 chunks of 96 so the chunk tile fits LDS
  for (int cc = 0; cc < 4; cc++) {
    // GEMM1: Hc = gelu(XN @ W1T[:, chunk] + b1)
    for (int j = 0; j < 3; j++) {
      int t = wave + j * 8;
      int mt = t & 3, ht = t >> 2;
      int cb = cc * 96 + ht * 16;
      v8f acc = {};
      for (int k = 0; k < 96; k += 4) {
        int kk = k + (hifrag ? 2 : 0);
        v2f a = *(const v2f*)&XN[(mt * 16 + lr) * XST + kk];
        v2f bv;
        bv.x = w1T[kk * 384 + cb + lr];
        bv.y = w1T[(kk + 1) * 384 + cb + lr];
        acc = wmma_f32(a, bv, acc);
      }
#pragma unroll
      for (int r = 0; r < 8; r++) {
        int row = mt * 16 + r + (hifrag ? 8 : 0);
        int col = ht * 16 + lr;
        Hc[row * XST + col] = gelu_exact(acc[r] + b1[cc * 96 + col]);
      }
    }
    __syncthreads();
    // GEMM2: outacc += Hc @ W2T[chunk, :]
    for (int j = 0; j < 3; j++) {
      int t = wave + j * 8;
      int mt = t & 3, nt = t >> 2;
      v8f acc = outacc[j];
      for (int k = 0; k < 96; k += 4) {
        int kk = k + (hifrag ? 2 : 0);
        v2f a = *(const v2f*)&Hc[(mt * 16 + lr) * XST + kk];
        v2f bv;
        bv.x = w2T[(cc * 96 + kk) * 96 + nt * 16 + lr];
        bv.y = w2T[(cc * 96 + kk + 1) * 96 + nt * 16 + lr];
        acc = wmma_f32(a, bv, acc);
      }
      outacc[j] = acc;
    }
    __syncthreads();
  }

  // write: + bias + residual
  for (int j = 0; j < 3; j++) {
    int t = wave + j * 8;
    int mt = t & 3, nt = t >> 2;
#pragma unroll
    for (int r = 0; r < 8; r++) {
      int row = mt * 16 + r + (hifrag ? 8 : 0);
      int col = nt * 16 + lr;
      size_t gi = (((size_t)bb * 96 + col) << 16) + sp + row;
      out[gi] = outacc[j][r] + b2[col] + xr[gi];
    }
  }
}

// ================================================================ launcher
extern "C" void kernel_launch(void* const* d_in, const int* in_sizes, int n_in,
                              void* d_out, int out_size, void* d_ws, size_t ws_size,
                              hipStream_t stream) {
  const float* x      = (const float*)d_in[0];
  const float* ln1_w  = (const float*)d_in[1];
  const float* ln1_b  = (const float*)d_in[2];
  const float* qkv_w  = (const float*)d_in[3];
  const float* qkv_b  = (const float*)d_in[4];
  const float* proj_w = (const float*)d_in[5];
  const float* proj_b = (const float*)d_in[6];
  const float* rpb    = (const float*)d_in[7];
  const float* cab1_w = (const float*)d_in[8];
  const float* cab1_b = (const float*)d_in[9];
  const float* cab2_w = (const float*)d_in[10];
  const float* cab2_b = (const float*)d_in[11];
  const float* ca1_w  = (const float*)d_in[12];
  const float* ca1_b  = (const float*)d_in[13];
  const float* ca2_w  = (const float*)d_in[14];
  const float* ca2_b  = (const float*)d_in[15];
  const float* ln2_w  = (const float*)d_in[16];
  const float* ln2_b  = (const float*)d_in[17];
  const float* mlp1_w = (const float*)d_in[18];
  const float* mlp1_b = (const float*)d_in[19];
  const float* mlp2_w = (const float*)d_in[20];
  const float* mlp2_b = (const float*)d_in[21];

  float* ws   = (float*)d_ws;
  float* U    = ws;                                   // u, later reused as x'
  float* MSA  = ws + (size_t)NTOT;
  float* C2   = ws + (size_t)2 * NTOT;
  float* C1   = ws + (size_t)3 * NTOT;                // NTOT/4 elements
  float* SM   = ws + (size_t)3 * NTOT + NTOT / 4;     // small buffers
  float* pool = SM;                                   // 192
  float* asc  = SM + 192;                             // 192
  float* qkvT = SM + 384;                             // 96*288
  float* projT = qkvT + 96 * 288;                     // 96*96
  float* m1T  = projT + 96 * 96;                      // 96*384
  float* m2T  = m1T + 96 * 384;                       // 384*96
  float* cw1T = m2T + 384 * 96;                       // 864*32
  float* cw2T = cw1T + 864 * 32;                      // 216*96

  k_transpose<<<(96 * 384 + 255) / 256, 256, 0, stream>>>(
      qkv_w, proj_w, mlp1_w, mlp2_w, cab1_w, cab2_w,
      qkvT, projT, m1T, m2T, cw1T, cw2T);
  k_ln<<<(2 * NPIX) / 256, 256, 0, stream>>>(x, ln1_w, ln1_b, U);
  k_swin<<<2048, 256, 0, stream>>>(U, qkvT, qkv_b, projT, proj_b, rpb, MSA);
  k_conv1<<<512, 256, 0, stream>>>(U, cw1T, cab1_b, C1);
  k_conv2<<<512, 256, 0, stream>>>(C1, cw2T, cab2_b, C2);
  k_pool<<<192, 256, 0, stream>>>(C2, pool);
  k_ca<<<2, 128, 0, stream>>>(pool, ca1_w, ca1_b, ca2_w, ca2_b, asc);
  k_combine<<<(NTOT + 255) / 256, 256, 0, stream>>>(x, MSA, C2, asc, U);
  k_mlp<<<2048, 256, 0, stream>>>(U, ln2_w, ln2_b, m1T, mlp1_b, m2T, mlp2_b,
                                  (float*)d_out);
}